// OpticalFlowShader_3418793968176
// MI455X (gfx1250) — compile-verified
//
#include <hip/hip_runtime.h>
#include <hip/hip_bf16.h>
#include <stdint.h>

// Problem constants (from the reference)
#define N_  16
#define H_  512
#define W_  512
#define V_  6890
#define F_  13776
#define HW_ (H_ * W_)          // 262144 = 2^18
#define TOTAL_PX_ (N_ * HW_)   // 4,194,304

typedef __attribute__((ext_vector_type(4))) unsigned int u32x4;
typedef __attribute__((ext_vector_type(8))) int          i32x8;
typedef __attribute__((ext_vector_type(4))) int          i32x4;

extern "C" __global__ __launch_bounds__(256)
void OpticalFlowShader_kernel(const int*   __restrict__ p2f,    // [N,H,W,1] int32
                              const float* __restrict__ bary,   // [N,H,W,1,3] f32
                              const int*   __restrict__ faces,  // [F,3] int32
                              const float* __restrict__ verts,  // [N,V,3] f32
                              float*       __restrict__ out,    // [N,H,W,3] f32
                              int total_px)
{
    extern __shared__ int lds_faces[]; // F_*3 = 41328 dwords (165,312 B) staged by TDM

    // ---------------- TDM: DMA the faces LUT into LDS (one op per workgroup) ----------------
    // TENSOR_LOAD_TO_LDS ignores EXEC but issues per-wave; gate to wave 0 so the
    // workgroup issues exactly one DMA. TENSORcnt is per-wave, so the issuing wave
    // waits, then the workgroup barrier publishes the LDS contents to all 8 waves.
    if (threadIdx.x < 32u) {
        const uint64_t ga      = (uint64_t)(uintptr_t)faces;
        const uint32_t lds_off = (uint32_t)(uintptr_t)lds_faces; // addr[31:0] == LDS byte offset
        const uint32_t ndw     = (uint32_t)(F_ * 3);             // 41328 dwords

        u32x4 g0;
        g0[0] = 1u;                                                // count=1, user-mode, no gather
        g0[1] = lds_off;                                           // D#.lds_addr (bytes)
        g0[2] = (uint32_t)(ga & 0xFFFFFFFFull);                    // global_addr[31:0]
        g0[3] = (uint32_t)((ga >> 32) & 0x1FFFFFFull) | (2u << 30);// global_addr[56:32] | type=2

        i32x8 g1;
        g1[0] = (int)(2u << 16);                                   // wg_mask=0, data_size=2 (4 B)
        g1[1] = (int)((ndw & 0xFFFFu) << 16);                      // tensor_dim0[15:0] @bits63:48
        g1[2] = (int)(((ndw >> 16) & 0xFFFFu) | (1u << 16));       // tensor_dim0[31:16], tensor_dim1=1
        g1[3] = (int)((ndw & 0xFFFFu) << 16);                      // tensor_dim1 hi=0, tile_dim0=ndw
        g1[4] = 1;                                                 // tile_dim1=1, tile_dim2=0
        g1[5] = (int)ndw;                                          // tensor_dim0_stride[31:0]
        g1[6] = 0;                                                 // stride hi / dim1_stride lo
        g1[7] = 0;

        i32x4 g2 = {0, 0, 0, 0};                                   // <=2D tensor: groups 2/3 unused
        i32x4 g3 = {0, 0, 0, 0};
        i32x8 g4 = {0, 0, 0, 0, 0, 0, 0, 0};                       // extra group (clang-23 6-arg form)

        __builtin_amdgcn_tensor_load_to_lds(g0, g1, g2, g3, g4, /*cpol=*/0);
        __builtin_amdgcn_s_wait_tensorcnt(0);
    }
    __syncthreads();

    // ---------------- Streaming main loop: 1 pixel per lane per iteration ----------------
    const float gstep = 2.0f / (float)(W_ - 1); // H_ == W_ so one step serves both axes
    const int   tid    = (int)(blockIdx.x * blockDim.x + threadIdx.x);
    const int   stride = (int)(gridDim.x * blockDim.x);

    for (int p = tid; p < total_px; p += stride) {
        // Prefetch the next grid-stride iteration's streamed operands (global_prefetch_b8)
        const int pn = p + stride;
        if (pn < total_px) {
            __builtin_prefetch(&bary[3 * pn], 0, 0);
            __builtin_prefetch(&p2f[pn],     0, 0);
        }

        const int   idx  = p2f[p];
        const float b0   = bary[3 * p + 0];
        const float b1   = bary[3 * p + 1];
        const float b2   = bary[3 * p + 2];

        const bool     valid = (idx >= 0);
        const unsigned fidx  = valid ? (unsigned)idx : 0u;
        const unsigned mesh  = fidx / (unsigned)F_;        // const-div -> mulhi sequence
        const unsigned local = fidx - mesh * (unsigned)F_;

        // Face vertex indices from the TDM-staged LDS LUT
        const int i0 = lds_faces[3 * local + 0];
        const int i1 = lds_faces[3 * local + 1];
        const int i2 = lds_faces[3 * local + 2];

        // Gather 3 vertices (L2-resident 1.3 MB table)
        const int base = (int)mesh * (V_ * 3);
        const float* __restrict__ va = verts + base + 3 * i0;
        const float* __restrict__ vb = verts + base + 3 * i1;
        const float* __restrict__ vc = verts + base + 3 * i2;

        float fx = fmaf(b0, va[0], fmaf(b1, vb[0], b2 * vc[0]));
        float fy = fmaf(b0, va[1], fmaf(b1, vb[1], b2 * vc[1]));
        float fz = fmaf(b0, va[2], fmaf(b1, vb[2], b2 * vc[2]));
        if (!valid) { fx = 0.0f; fy = 0.0f; fz = 0.0f; }

        // Analytic mesh grid: gx over W, gy over H, gz = 0
        const int hw = p & (HW_ - 1);
        const int h  = hw >> 9;         // W_ = 512
        const int w  = hw & (W_ - 1);

        out[3 * p + 0] = fx + fmaf((float)w, gstep, -1.0f);
        out[3 * p + 1] = fy + fmaf((float)h, gstep, -1.0f);
        out[3 * p + 2] = fz;
    }
}

extern "C" void kernel_launch(void* const* d_in, const int* in_sizes, int n_in,
                              void* d_out, int out_size, void* d_ws, size_t ws_size,
                              hipStream_t stream) {
    const int*   p2f   = (const int*)  d_in[0];  // pix_to_face
    const float* bary  = (const float*)d_in[1];  // bary_coords
    const int*   faces = (const int*)  d_in[2];  // faces
    const float* verts = (const float*)d_in[3];  // verts_scene_flow
    float*       out   = (float*)d_out;

    const size_t lds_bytes = (size_t)F_ * 3 * sizeof(int); // 165,312 B (<= 320 KB/WGP)
    // Raise the dynamic-LDS cap (not a stream op; deterministic, touches no buffers).
    (void)hipFuncSetAttribute((const void*)OpticalFlowShader_kernel,
                              hipFuncAttributeMaxDynamicSharedMemorySize, (int)lds_bytes);

    dim3 block(256);
    dim3 grid(512); // 32 px/thread grid-stride; keeps redundant TDM staging ~85 MB of L2 traffic

    OpticalFlowShader_kernel<<<grid, block, lds_bytes, stream>>>(
        p2f, bary, faces, verts, out, TOTAL_PX_);
}